// Net_85426899517808
// MI455X (gfx1250) — compile-verified
//
#include <hip/hip_runtime.h>
#include <math.h>

typedef __attribute__((ext_vector_type(2))) float v2f;
typedef __attribute__((ext_vector_type(4))) float v4f;
typedef __attribute__((ext_vector_type(8))) float v8f;

// ---------------------------------------------------------------------------
// Degree count: deg[row[e]] += 1 over the 3.2M explicit edges.
// ---------------------------------------------------------------------------
__global__ void k_deg_count(const int* __restrict__ row, float* __restrict__ deg, int E) {
    int i = blockIdx.x * blockDim.x + threadIdx.x;
    if (i < E) atomicAdd(&deg[row[i]], 1.0f);
}

// dis[i] = (deg[i] + 1)^-0.5   (+1 = self loop)
__global__ void k_deg_to_dis(float* __restrict__ deg, int n) {
    int i = blockIdx.x * blockDim.x + threadIdx.x;
    if (i < n) deg[i] = rsqrtf(deg[i] + 1.0f);
}

// ---------------------------------------------------------------------------
// GEMM1: h1[N,16] = x[N,512] @ W1[512,16] + b1, via V_WMMA_F32_16X16X4_F32.
// Block = 256 threads = 8 waves, 128 rows per block. x staged in LDS in
// 128x64 chunks (fully coalesced float4 loads, padded stride 68 to avoid
// 16-way bank conflicts on column reads).
//
// fp32 WMMA fragment layout (ISA 7.12.2):
//   A 16x4:  lane L -> row m=L%16;  VGPR0 = K=(L<16?0:2), VGPR1 = K+1
//   B 4x16:  lane L -> col n=L%16;  VGPR0 = row K=(L<16?0:2), VGPR1 = K+1
//   C 16x16: VGPR j: lanes 0-15 -> M=j, N=lane; lanes 16-31 -> M=j+8, N=lane-16
// ---------------------------------------------------------------------------
__global__ __launch_bounds__(256) void k_gemm1_wmma(
    const float* __restrict__ x, const float* __restrict__ W1,
    const float* __restrict__ b1, float* __restrict__ h1, int nrows) {
    __shared__ float xs[128 * 68];   // 34.8 KB, padded stride
    __shared__ float wsm[64 * 16];   // 4 KB W1 chunk

    const int t    = threadIdx.x;
    const int lane = t & 31;
    const int wave = t >> 5;
    const int row0 = blockIdx.x * 128;
    const int m16  = lane & 15;
    const int kb   = (lane < 16) ? 0 : 2;

    v8f acc = {};

    for (int k0 = 0; k0 < 512; k0 += 64) {
        __syncthreads();
        // stage x[row0..row0+127][k0..k0+63]: 2048 float4, 8 per thread
        #pragma unroll
        for (int i = 0; i < 8; ++i) {
            int idx  = t + i * 256;          // float4 index
            int r    = idx >> 4;             // 0..127
            int c4   = idx & 15;             // 0..15
            int grow = row0 + r;
            v4f val = {};
            if (grow < nrows)
                val = *(const v4f*)(x + (size_t)grow * 512 + k0 + c4 * 4);
            float* dst = &xs[r * 68 + c4 * 4];
            dst[0] = val.x; dst[1] = val.y; dst[2] = val.z; dst[3] = val.w;
        }
        // stage W1[k0..k0+63][0..15]: 256 float4, 1 per thread
        {
            int r  = t >> 2;                 // 0..63
            int c4 = t & 3;
            v4f wv = *(const v4f*)(W1 + (size_t)(k0 + r) * 16 + c4 * 4);
            float* dst = &wsm[r * 16 + c4 * 4];
            dst[0] = wv.x; dst[1] = wv.y; dst[2] = wv.z; dst[3] = wv.w;
        }
        __syncthreads();

        const float* arow = &xs[(wave * 16 + m16) * 68];
        #pragma unroll
        for (int kk = 0; kk < 64; kk += 4) {
            v2f a = *(const v2f*)(arow + kk + kb);           // ds_load_b64
            v2f b;
            b.x = wsm[(kk + kb)     * 16 + m16];
            b.y = wsm[(kk + kb + 1) * 16 + m16];
            acc = __builtin_amdgcn_wmma_f32_16x16x4_f32(
                false, a, false, b, (short)0, acc, false, false);
        }
    }

    const int mbase = (lane < 16) ? 0 : 8;
    const int rowb  = row0 + wave * 16;
    const float bias = b1[m16];
    #pragma unroll
    for (int j = 0; j < 8; ++j) {
        int grow = rowb + mbase + j;
        if (grow < nrows)
            h1[(size_t)grow * 16 + m16] = acc[j] + bias;
    }
}

// ---------------------------------------------------------------------------
// GEMM2: h2[N,40] = relu(agg1[N,16]) @ W2[16,40] + b2. W2 zero-padded to
// 16x48 in LDS; 3 N-tiles x (K=16 -> 4 WMMA steps) = 12 WMMAs per wave.
// ReLU fused into the A-fragment load.
// ---------------------------------------------------------------------------
__global__ __launch_bounds__(256) void k_gemm2_wmma(
    const float* __restrict__ agg1, const float* __restrict__ W2,
    const float* __restrict__ b2, float* __restrict__ h2, int nrows) {
    __shared__ float wsm[16 * 48];

    const int t = threadIdx.x;
    for (int i = t; i < 16 * 48; i += 256) {
        int r = i / 48, c = i % 48;
        wsm[i] = (c < 40) ? W2[r * 40 + c] : 0.0f;
    }
    __syncthreads();

    const int lane = t & 31;
    const int wave = t >> 5;
    const int rowb = (blockIdx.x * 8 + wave) * 16;
    if (rowb >= nrows) return;   // wave-uniform: 100000 % 16 == 0

    const int m16 = lane & 15;
    const int kb  = (lane < 16) ? 0 : 2;

    v8f acc0 = {}, acc1 = {}, acc2 = {};
    const float* arow = agg1 + (size_t)(rowb + m16) * 16;

    #pragma unroll
    for (int k = 0; k < 16; k += 4) {
        v2f araw = *(const v2f*)(arow + k + kb);
        v2f a; a.x = fmaxf(araw.x, 0.0f); a.y = fmaxf(araw.y, 0.0f);  // ReLU
        const float* w0 = &wsm[(k + kb)     * 48];
        const float* w1 = &wsm[(k + kb + 1) * 48];
        v2f b0; b0.x = w0[m16];      b0.y = w1[m16];
        v2f b1; b1.x = w0[16 + m16]; b1.y = w1[16 + m16];
        v2f b2v; b2v.x = w0[32 + m16]; b2v.y = w1[32 + m16];
        acc0 = __builtin_amdgcn_wmma_f32_16x16x4_f32(false, a, false, b0,  (short)0, acc0, false, false);
        acc1 = __builtin_amdgcn_wmma_f32_16x16x4_f32(false, a, false, b1,  (short)0, acc1, false, false);
        acc2 = __builtin_amdgcn_wmma_f32_16x16x4_f32(false, a, false, b2v, (short)0, acc2, false, false);
    }

    const int mbase = (lane < 16) ? 0 : 8;
    #pragma unroll
    for (int j = 0; j < 8; ++j) {
        int m = rowb + mbase + j;
        if (m < nrows) {
            float* orow = h2 + (size_t)m * 40;
            orow[m16]      = acc0[j] + b2[m16];
            orow[16 + m16] = acc1[j] + b2[16 + m16];
            if (32 + m16 < 40) orow[32 + m16] = acc2[j] + b2[32 + m16];
        }
    }
}

// ---------------------------------------------------------------------------
// Edge aggregation: out[col[e]*F + f] += dis[row]*dis[col] * h[row*F + f].
// One thread per (edge, feature); self-loops synthesized for e >= E.
// h and out stay L2-resident (6.4 / 16 MB vs 192 MB L2).
// ---------------------------------------------------------------------------
template <int F>
__global__ void k_aggregate(const float* __restrict__ h,
                            const int* __restrict__ row, const int* __restrict__ col,
                            const float* __restrict__ dis, float* __restrict__ out,
                            int E, unsigned total) {
    unsigned idx = blockIdx.x * blockDim.x + threadIdx.x;
    if (idx >= total) return;
    int e = idx / F;
    int f = idx % F;
    int r, c;
    if (e < E) { r = row[e]; c = col[e]; }
    else       { r = c = e - E; }          // self loop
    float norm = dis[r] * dis[c];
    atomicAdd(&out[(size_t)c * F + f], norm * h[(size_t)r * F + f]);
}

// ---------------------------------------------------------------------------
// In-place row-wise log_softmax over 40 classes. One thread owns one row.
// ---------------------------------------------------------------------------
__global__ void k_log_softmax40(float* __restrict__ out, int n) {
    int i = blockIdx.x * blockDim.x + threadIdx.x;
    if (i >= n) return;
    float* rowp = out + (size_t)i * 40;
    float mx = -INFINITY;
    #pragma unroll
    for (int j = 0; j < 40; ++j) mx = fmaxf(mx, rowp[j]);
    float s = 0.0f;
    #pragma unroll
    for (int j = 0; j < 40; ++j) s += expf(rowp[j] - mx);
    float ls = logf(s);
    #pragma unroll
    for (int j = 0; j < 40; ++j) rowp[j] = rowp[j] - mx - ls;
}

// ---------------------------------------------------------------------------
extern "C" void kernel_launch(void* const* d_in, const int* in_sizes, int n_in,
                              void* d_out, int out_size, void* d_ws, size_t ws_size,
                              hipStream_t stream) {
    const float* x    = (const float*)d_in[0];
    const int*   eidx = (const int*)  d_in[1];
    const float* W1   = (const float*)d_in[2];
    const float* b1   = (const float*)d_in[3];
    const float* W2   = (const float*)d_in[4];
    const float* b2   = (const float*)d_in[5];
    float* out = (float*)d_out;

    const int N = in_sizes[0] / 512;   // 100000
    const int E = in_sizes[1] / 2;     // 3200000
    const int* row = eidx;             // edge_index[0]
    const int* col = eidx + E;         // edge_index[1]

    // workspace layout (floats), 256-float aligned
    auto align = [](size_t v) { return (v + 255) & ~(size_t)255; };
    float* ws  = (float*)d_ws;
    size_t o = 0;
    float* dis  = ws + o; o += align((size_t)N);
    float* h1   = ws + o; o += align((size_t)N * 16);
    float* agg1 = ws + o; o += align((size_t)N * 16);
    float* h2   = ws + o; o += align((size_t)N * 40);
    (void)ws_size;

    // per-call re-init of everything accumulated into
    hipMemsetAsync(dis,  0, (size_t)N * sizeof(float),      stream);
    hipMemsetAsync(agg1, 0, (size_t)N * 16 * sizeof(float), stream);
    hipMemsetAsync(out,  0, (size_t)N * 40 * sizeof(float), stream);

    const int B = 256;
    // degree -> dis
    k_deg_count<<<(E + B - 1) / B, B, 0, stream>>>(row, dis, E);
    k_deg_to_dis<<<(N + B - 1) / B, B, 0, stream>>>(dis, N);

    // layer 1
    int tiles  = (N + 15) / 16;
    int blocks = (tiles + 7) / 8;
    k_gemm1_wmma<<<blocks, B, 0, stream>>>(x, W1, b1, h1, N);
    unsigned tot1 = (unsigned)(E + N) * 16u;
    k_aggregate<16><<<(tot1 + B - 1) / B, B, 0, stream>>>(h1, row, col, dis, agg1, E, tot1);

    // layer 2 (ReLU fused into GEMM2 A-load)
    k_gemm2_wmma<<<blocks, B, 0, stream>>>(agg1, W2, b2, h2, N);
    unsigned tot2 = (unsigned)(E + N) * 40u;
    k_aggregate<40><<<(tot2 + B - 1) / B, B, 0, stream>>>(h2, row, col, dis, out, E, tot2);

    // log_softmax in place on d_out
    k_log_softmax40<<<(N + B - 1) / B, B, 0, stream>>>(out, N);
}